// WaterMaskHead_55989193671077
// MI455X (gfx1250) — compile-verified
//
#include <hip/hip_runtime.h>
#include <hip/hip_bf16.h>
#include <math.h>

// ---------------------------------------------------------------------------
// WaterMaskHead pipeline for gfx1250 (CDNA5), fp32 WMMA path with
// double-buffered async global->LDS staging (global_load_async_to_lds_b32/b128).
// B=32, C=256, H=W=96, tok=3 -> tokens N=1024/batch, Fh=256, TOP_K=11.
// Both A and B tiles are stored K-contiguous per row in LDS (stride 36 dwords,
// conflict-free), so every WMMA fragment is a single 8-byte LDS load.
// ---------------------------------------------------------------------------

typedef float v2f __attribute__((ext_vector_type(2)));
typedef float v8f __attribute__((ext_vector_type(8)));

#define NEG_INF (-INFINITY)

// ----- CDNA5 async global->LDS helpers (inline asm, ASYNCcnt-tracked) ------
__device__ __forceinline__ unsigned lds_off(const void* p) {
    // generic pointer to LDS: low 32 bits are the workgroup-relative offset
    return (unsigned)(size_t)p;
}
__device__ __forceinline__ void async_ld_b32(unsigned lds, const void* base,
                                             unsigned byte_off) {
    asm volatile("global_load_async_to_lds_b32 %0, %1, %2"
                 :: "v"(lds), "v"(byte_off),
                    "s"((unsigned long long)(size_t)base)
                 : "memory");
}
__device__ __forceinline__ void async_ld_b128(unsigned lds, const void* base,
                                              unsigned byte_off) {
    asm volatile("global_load_async_to_lds_b128 %0, %1, %2"
                 :: "v"(lds), "v"(byte_off),
                    "s"((unsigned long long)(size_t)base)
                 : "memory");
}
__device__ __forceinline__ void wait_async() {
    asm volatile("s_wait_asynccnt 0" ::: "memory");
}

// ----- fp32 WMMA 16x16x4 wrapper -------------------------------------------
__device__ __forceinline__ v8f wmma4(v2f a, v2f b, v8f c) {
    return __builtin_amdgcn_wmma_f32_16x16x4_f32(false, a, false, b,
                                                 (short)0, c, false, false);
}

// ----- 64x64 block tile, 4 waves, each wave owns 32x32 (2x2 register block).
// As[m][k], Bt[n][k]; row stride 36 dwords: banks 36*r mod 64 are distinct for
// r=0..15 and the two half-wave groups never collide -> conflict-free b64 reads.
__device__ __forceinline__ void tile_mma(const float (*As)[36], const float (*Bt)[36],
                                         int wi, int wj, int half, int lr,
                                         v8f acc[2][2]) {
#pragma unroll
    for (int kk = 0; kk < 32; kk += 4) {
        const int kc = kk + 2 * half;                 // even -> 8B aligned
        v2f a0 = *(const v2f*)&As[32 * wi + lr][kc];
        v2f a1 = *(const v2f*)&As[32 * wi + 16 + lr][kc];
        v2f b0 = *(const v2f*)&Bt[32 * wj + lr][kc];
        v2f b1 = *(const v2f*)&Bt[32 * wj + 16 + lr][kc];
        acc[0][0] = wmma4(a0, b0, acc[0][0]);
        acc[0][1] = wmma4(a0, b1, acc[0][1]);
        acc[1][0] = wmma4(a1, b0, acc[1][0]);
        acc[1][1] = wmma4(a1, b1, acc[1][1]);
    }
}

// ---------------------------------------------------------------------------
// Kernel 1: patchify conv as GEMM.  M=32768 (b,py,px), K=2304 (c,ky,kx), N=256.
// Epilogue: +bias, ReLU.  grid (4, 512), 128 threads.
// A is a gather (per-dword async); B rows cw[n][k] are k-contiguous (b128).
// ---------------------------------------------------------------------------
__global__ __launch_bounds__(128)
void k_conv_patchify(const float* __restrict__ x, const float* __restrict__ cw,
                     const float* __restrict__ cb, float* __restrict__ h) {
    __shared__ __align__(16) float As[2][64][36];
    __shared__ __align__(16) float Bt[2][64][36];
    const int tid = threadIdx.x;
    const int wave = tid >> 5, lane = tid & 31;
    const int half = lane >> 4, lr = lane & 15;
    const int wi = wave >> 1, wj = wave & 1;
    const int mbase = blockIdx.y * 64;
    const int nbase = blockIdx.x * 64;

    auto stage = [&](int buf, int kbase) {
        for (int t = tid; t < 2048; t += 128) {           // A: 64 rows x 32 k
            int r = t >> 5, k = t & 31;
            int m = mbase + r, kk = kbase + k;
            int b = m >> 10, p = m & 1023, py = p >> 5, px = p & 31;
            int c = kk / 9, rem = kk - 9 * c, ky = rem / 3, kx = rem - 3 * ky;
            unsigned go = (unsigned)((((b * 256 + c) * 96 + 3 * py + ky) * 96
                                      + 3 * px + kx) * 4);
            async_ld_b32(lds_off(&As[buf][r][k]), x, go);
        }
        for (int t = tid; t < 512; t += 128) {            // Bt[n][k] = cw[n][k]
            int n = t >> 3, k4 = (t & 7) * 4;
            unsigned go = (unsigned)(((nbase + n) * 2304 + kbase + k4) * 4);
            async_ld_b128(lds_off(&Bt[buf][n][k4]), cw, go);
        }
    };

    v8f acc[2][2] = {};
    stage(0, 0);
    wait_async();
    __syncthreads();
    const int NC = 72;                                    // 2304 / 32
    for (int c = 0; c < NC; ++c) {
        int cur = c & 1;
        if (c + 1 < NC) stage(cur ^ 1, (c + 1) * 32);
        tile_mma(As[cur], Bt[cur], wi, wj, half, lr, acc);
        wait_async();
        __syncthreads();
    }
#pragma unroll
    for (int bi = 0; bi < 2; ++bi) {
        const int col = nbase + 32 * wj + 16 * bi + lr;
        const float bias = cb[col];
#pragma unroll
        for (int ai = 0; ai < 2; ++ai)
#pragma unroll
            for (int v = 0; v < 8; ++v) {
                int m = mbase + 32 * wi + 16 * ai + v + 8 * half;
                h[(size_t)m * 256 + col] = fmaxf(acc[ai][bi][v] + bias, 0.f);
            }
    }
}

// ---------------------------------------------------------------------------
// Kernel 2: per-token squared norms (one wave per row of 256).
// ---------------------------------------------------------------------------
__global__ void k_row_sq(const float* __restrict__ h, float* __restrict__ sq) {
    int row  = blockIdx.x * 8 + (threadIdx.x >> 5);
    int lane = threadIdx.x & 31;
    const float* hr = h + (size_t)row * 256;
    float s = 0.f;
    for (int k = lane; k < 256; k += 32) { float v = hr[k]; s += v * v; }
    for (int off = 16; off > 0; off >>= 1) s += __shfl_down(s, off);
    if (lane == 0) sq[row] = s;
}

// ---------------------------------------------------------------------------
// Kernel 3: pairwise distances per batch: S = sqrt(max(0, sq_i+sq_j-2*h@h^T)).
// grid (16, 16, 32 batches).  Both tiles k-contiguous -> b128 staging.
// ---------------------------------------------------------------------------
__global__ __launch_bounds__(128)
void k_dist(const float* __restrict__ h, const float* __restrict__ sq,
            float* __restrict__ S) {
    __shared__ __align__(16) float As[2][64][36];
    __shared__ __align__(16) float Bt[2][64][36];
    const int tid = threadIdx.x;
    const int wave = tid >> 5, lane = tid & 31;
    const int half = lane >> 4, lr = lane & 15;
    const int wi = wave >> 1, wj = wave & 1;
    const int bz = blockIdx.z;
    const int ibase = blockIdx.y * 64;
    const int jbase = blockIdx.x * 64;
    const float* hb = h + (size_t)bz * 1024 * 256;

    auto stage = [&](int buf, int kbase) {
        for (int t = tid; t < 512; t += 128) {
            int r = t >> 3, k4 = (t & 7) * 4;
            async_ld_b128(lds_off(&As[buf][r][k4]), hb,
                          (unsigned)(((ibase + r) * 256 + kbase + k4) * 4));
        }
        for (int t = tid; t < 512; t += 128) {
            int j = t >> 3, k4 = (t & 7) * 4;
            async_ld_b128(lds_off(&Bt[buf][j][k4]), hb,
                          (unsigned)(((jbase + j) * 256 + kbase + k4) * 4));
        }
    };

    v8f acc[2][2] = {};
    stage(0, 0);
    wait_async();
    __syncthreads();
    const int NC = 8;                                     // 256 / 32
    for (int c = 0; c < NC; ++c) {
        int cur = c & 1;
        if (c + 1 < NC) stage(cur ^ 1, (c + 1) * 32);
        tile_mma(As[cur], Bt[cur], wi, wj, half, lr, acc);
        wait_async();
        __syncthreads();
    }
#pragma unroll
    for (int bi = 0; bi < 2; ++bi) {
        int j = jbase + 32 * wj + 16 * bi + lr;
        float sqj = sq[bz * 1024 + j];
#pragma unroll
        for (int ai = 0; ai < 2; ++ai)
#pragma unroll
            for (int v = 0; v < 8; ++v) {
                int i = ibase + 32 * wi + 16 * ai + v + 8 * half;
                float d2 = sq[bz * 1024 + i] + sqj - 2.f * acc[ai][bi][v];
                S[((size_t)bz * 1024 + i) * 1024 + j] = sqrtf(fmaxf(d2, 0.f));
            }
    }
}

// ---------------------------------------------------------------------------
// Kernel 4: top-11 largest per row (ties -> smallest index, like lax.top_k).
// ---------------------------------------------------------------------------
__global__ void k_topk(const float* __restrict__ S, int* __restrict__ kidx) {
    __shared__ float row[1024];
    __shared__ float rv[256];
    __shared__ int   ri[256];
    const int r = blockIdx.x;                  // b*1024 + i
    const int tid = threadIdx.x;
    const float* Sr = S + (size_t)r * 1024;
    for (int j = tid; j < 1024; j += 256) row[j] = Sr[j];
    __syncthreads();
    for (int t = 0; t < 11; ++t) {
        float bv = NEG_INF; int bi = 0x7fffffff;
        for (int j = tid; j < 1024; j += 256) {
            float v = row[j];
            if (v > bv || (v == bv && j < bi)) { bv = v; bi = j; }
        }
        rv[tid] = bv; ri[tid] = bi;
        __syncthreads();
        for (int s = 128; s > 0; s >>= 1) {
            if (tid < s) {
                float ov = rv[tid + s]; int oi = ri[tid + s];
                if (ov > rv[tid] || (ov == rv[tid] && oi < ri[tid])) {
                    rv[tid] = ov; ri[tid] = oi;
                }
            }
            __syncthreads();
        }
        if (tid == 0) { kidx[(size_t)r * 11 + t] = ri[0]; row[ri[0]] = NEG_INF; }
        __syncthreads();
    }
}

// ---------------------------------------------------------------------------
// Kernel 5: Wh = hin @ W  (M=32768, K=256, N=256).  grid (4, 512).
// A k-contiguous (b128); W is n-contiguous so B stages per-dword (b32).
// ---------------------------------------------------------------------------
__global__ __launch_bounds__(128)
void k_feat_gemm(const float* __restrict__ hin, const float* __restrict__ W,
                 float* __restrict__ Wh) {
    __shared__ __align__(16) float As[2][64][36];
    __shared__ __align__(16) float Bt[2][64][36];
    const int tid = threadIdx.x;
    const int wave = tid >> 5, lane = tid & 31;
    const int half = lane >> 4, lr = lane & 15;
    const int wi = wave >> 1, wj = wave & 1;
    const int mbase = blockIdx.y * 64;
    const int nbase = blockIdx.x * 64;

    auto stage = [&](int buf, int kbase) {
        for (int t = tid; t < 512; t += 128) {
            int r = t >> 3, k4 = (t & 7) * 4;
            async_ld_b128(lds_off(&As[buf][r][k4]), hin,
                          (unsigned)(((mbase + r) * 256 + kbase + k4) * 4));
        }
        for (int t = tid; t < 2048; t += 128) {
            int k = t >> 6, n = t & 63;                  // lanes walk n: coalesced
            async_ld_b32(lds_off(&Bt[buf][n][k]), W,
                         (unsigned)(((kbase + k) * 256 + nbase + n) * 4));
        }
    };

    v8f acc[2][2] = {};
    stage(0, 0);
    wait_async();
    __syncthreads();
    const int NC = 8;
    for (int c = 0; c < NC; ++c) {
        int cur = c & 1;
        if (c + 1 < NC) stage(cur ^ 1, (c + 1) * 32);
        tile_mma(As[cur], Bt[cur], wi, wj, half, lr, acc);
        wait_async();
        __syncthreads();
    }
#pragma unroll
    for (int bi = 0; bi < 2; ++bi) {
        const int col = nbase + 32 * wj + 16 * bi + lr;
#pragma unroll
        for (int ai = 0; ai < 2; ++ai)
#pragma unroll
            for (int v = 0; v < 8; ++v) {
                int m = mbase + 32 * wi + 16 * ai + v + 8 * half;
                Wh[(size_t)m * 256 + col] = acc[ai][bi][v];
            }
    }
}

// ---------------------------------------------------------------------------
// Kernel 6: Wh1 = Wh@a[:256], Wh2 = Wh@a[256:].  One wave per row.
// ---------------------------------------------------------------------------
__global__ void k_row_dot2(const float* __restrict__ Wh, const float* __restrict__ a,
                           float* __restrict__ Wh1, float* __restrict__ Wh2) {
    int row  = blockIdx.x * 8 + (threadIdx.x >> 5);
    int lane = threadIdx.x & 31;
    const float* wr = Wh + (size_t)row * 256;
    float s1 = 0.f, s2 = 0.f;
    for (int k = lane; k < 256; k += 32) {
        float v = wr[k];
        s1 += v * a[k];
        s2 += v * a[256 + k];
    }
    for (int off = 16; off > 0; off >>= 1) {
        s1 += __shfl_down(s1, off);
        s2 += __shfl_down(s2, off);
    }
    if (lane == 0) { Wh1[row] = s1; Wh2[row] = s2; }
}

// ---------------------------------------------------------------------------
// Kernel 7: exact sparse attention aggregate + ELU.
// Masked softmax weights are exactly 0 in fp32 (exp(-9e15 - max) underflows),
// so only the 11 top-k neighbors contribute: h'_i = sum_t w_t * Wh[j_t].
// ---------------------------------------------------------------------------
__global__ void k_gat_aggregate(const float* __restrict__ Wh,
                                const float* __restrict__ Wh1,
                                const float* __restrict__ Wh2,
                                const int* __restrict__ kidx,
                                float* __restrict__ hout) {
    __shared__ float w[11];
    __shared__ int   js[11];
    const int r = blockIdx.x;                  // b*1024 + i
    const int b = r >> 10;
    const int tid = threadIdx.x;
    if (tid == 0) {
        float e[11];
        float m = NEG_INF;
        const float w1 = Wh1[r];
        for (int t = 0; t < 11; ++t) {
            int j = kidx[(size_t)r * 11 + t];
            js[t] = b * 1024 + j;
            float xv = w1 + Wh2[b * 1024 + j];
            xv = xv > 0.f ? xv : 0.2f * xv;    // leaky_relu(., 0.2)
            e[t] = xv;
            m = fmaxf(m, xv);
        }
        float s = 0.f;
        for (int t = 0; t < 11; ++t) { e[t] = expf(e[t] - m); s += e[t]; }
        float inv = 1.f / s;
        for (int t = 0; t < 11; ++t) w[t] = e[t] * inv;
    }
    __syncthreads();
    for (int f = tid; f < 256; f += blockDim.x) {
        float acc = 0.f;
#pragma unroll
        for (int t = 0; t < 11; ++t) acc += w[t] * Wh[(size_t)js[t] * 256 + f];
        hout[(size_t)r * 256 + f] = acc > 0.f ? acc : (expf(acc) - 1.f);  // ELU
    }
}

// ---------------------------------------------------------------------------
// Kernel 8: deconv (kernel=stride=3 -> no overlap) as GEMM with scatter
// epilogue; fuses bias, residual add and final ReLU into d_out.
// M=32768 tokens, K=256, N=2304 (o,ky,kx).  grid (36, 512).
// ---------------------------------------------------------------------------
__global__ __launch_bounds__(128)
void k_deconv(const float* __restrict__ h2, const float* __restrict__ dw,
              const float* __restrict__ db, const float* __restrict__ x,
              float* __restrict__ out) {
    __shared__ __align__(16) float As[2][64][36];
    __shared__ __align__(16) float Bt[2][64][36];
    const int tid = threadIdx.x;
    const int wave = tid >> 5, lane = tid & 31;
    const int half = lane >> 4, lr = lane & 15;
    const int wi = wave >> 1, wj = wave & 1;
    const int mbase = blockIdx.y * 64;
    const int nbase = blockIdx.x * 64;

    auto stage = [&](int buf, int kbase) {
        for (int t = tid; t < 512; t += 128) {
            int r = t >> 3, k4 = (t & 7) * 4;
            async_ld_b128(lds_off(&As[buf][r][k4]), h2,
                          (unsigned)(((mbase + r) * 256 + kbase + k4) * 4));
        }
        for (int t = tid; t < 2048; t += 128) {
            int k = t >> 6, n = t & 63;                  // lanes walk n: coalesced
            async_ld_b32(lds_off(&Bt[buf][n][k]), dw,
                         (unsigned)(((kbase + k) * 2304 + nbase + n) * 4));
        }
    };

    v8f acc[2][2] = {};
    stage(0, 0);
    wait_async();
    __syncthreads();
    const int NC = 8;
    for (int c = 0; c < NC; ++c) {
        int cur = c & 1;
        if (c + 1 < NC) stage(cur ^ 1, (c + 1) * 32);
        tile_mma(As[cur], Bt[cur], wi, wj, half, lr, acc);
        wait_async();
        __syncthreads();
    }
#pragma unroll
    for (int bi = 0; bi < 2; ++bi) {
        int n = nbase + 32 * wj + 16 * bi + lr;
        int o = n / 9, rem = n - 9 * o, ky = rem / 3, kx = rem - 3 * ky;
        float bias = db[o];
#pragma unroll
        for (int ai = 0; ai < 2; ++ai)
#pragma unroll
            for (int v = 0; v < 8; ++v) {
                int m = mbase + 32 * wi + 16 * ai + v + 8 * half;
                int b = m >> 10, p = m & 1023, py = p >> 5, px = p & 31;
                size_t idx = ((size_t)(b * 256 + o) * 96 + 3 * py + ky) * 96
                             + 3 * px + kx;
                out[idx] = fmaxf(acc[ai][bi][v] + bias + x[idx], 0.f);
            }
    }
}

// ---------------------------------------------------------------------------
// Host-side launch sequence (graph-capture safe: only kernel launches).
// ---------------------------------------------------------------------------
extern "C" void kernel_launch(void* const* d_in, const int* in_sizes, int n_in,
                              void* d_out, int out_size, void* d_ws, size_t ws_size,
                              hipStream_t stream) {
    const float* x        = (const float*)d_in[0];   // (32,256,96,96)
    const float* conv_w   = (const float*)d_in[1];   // (256,256,3,3)
    const float* conv_b   = (const float*)d_in[2];   // (256,)
    const float* W_att    = (const float*)d_in[3];   // (256,256)
    const float* a_att    = (const float*)d_in[4];   // (512,1)
    const float* W_out    = (const float*)d_in[5];   // (256,256)
    const float* a_out    = (const float*)d_in[6];   // (512,1)
    const float* deconv_w = (const float*)d_in[7];   // (256,256,3,3)
    const float* deconv_b = (const float*)d_in[8];   // (256,)
    float* out = (float*)d_out;

    // Workspace layout (~237 MB)
    const size_t TOK = (size_t)32 * 1024 * 256 * sizeof(float);   // 33,554,432
    char* ws = (char*)d_ws;
    float* h   = (float*)(ws);
    float* Wh  = (float*)(ws + TOK);
    float* h1  = (float*)(ws + 2 * TOK);
    float* S   = (float*)(ws + 3 * TOK);                          // 134,217,728
    char*  tail = ws + 3 * TOK + (size_t)32 * 1024 * 1024 * sizeof(float);
    float* sq  = (float*)tail;
    float* Wh1 = sq + 32768;
    float* Wh2 = Wh1 + 32768;
    int*   kidx = (int*)(Wh2 + 32768);
    float* h2 = h;   // h is dead after layer-1's Wh; reuse it for h2

    dim3 blk(128);
    // 1. patchify conv + ReLU  -> h (B,1024,256)
    k_conv_patchify<<<dim3(4, 512), blk, 0, stream>>>(x, conv_w, conv_b, h);
    // 2. pairwise distances
    k_row_sq<<<4096, 256, 0, stream>>>(h, sq);
    k_dist<<<dim3(16, 16, 32), blk, 0, stream>>>(h, sq, S);
    // 3. top-11 largest distances -> neighbor indices
    k_topk<<<32768, 256, 0, stream>>>(S, kidx);
    // 4. GAT layer 1 (ELU)
    k_feat_gemm<<<dim3(4, 512), blk, 0, stream>>>(h, W_att, Wh);
    k_row_dot2<<<4096, 256, 0, stream>>>(Wh, a_att, Wh1, Wh2);
    k_gat_aggregate<<<32768, 128, 0, stream>>>(Wh, Wh1, Wh2, kidx, h1);
    // 5. GAT layer 2 (ELU)
    k_feat_gemm<<<dim3(4, 512), blk, 0, stream>>>(h1, W_out, Wh);
    k_row_dot2<<<4096, 256, 0, stream>>>(Wh, a_out, Wh1, Wh2);
    k_gat_aggregate<<<32768, 128, 0, stream>>>(Wh, Wh1, Wh2, kidx, h2);
    // 6. deconv + bias + residual + ReLU  (bilinear is identity at 96x96)
    k_deconv<<<dim3(36, 512), blk, 0, stream>>>(h2, deconv_w, deconv_b, x, out);
}